// QNNModel_11587821765297
// MI455X (gfx1250) — compile-verified
//
#include <hip/hip_runtime.h>
#include <math.h>

typedef float v2f __attribute__((ext_vector_type(2)));
typedef float v8f __attribute__((ext_vector_type(8)));

#define NQ   6
#define DIM  64
#define BS   128   // samples (threads) per block; 4 waves

// ------------------------------------------------------------------
// Prologue: build the fixed 64x64 complex unitary U (two variational
// layers + CZ masks) applied after the encoding layer.
// Thread j computes U|e_j> and stores row j of U^T into ws:
//   Mr[j*64 + i] = Re(U[i][j]),  Mi[j*64 + i] = Im(U[i][j])
// so that psi[b][i] = sum_j E[b][j] * M[j][i].
// ------------------------------------------------------------------

template<int Q>
__device__ __forceinline__ void apply_rx(float* sr, float* si, float c, float s) {
  constexpr int R = 1 << (5 - Q);
  constexpr int L = 1 << Q;
#pragma unroll
  for (int l = 0; l < L; ++l) {
#pragma unroll
    for (int r = 0; r < R; ++r) {
      const int i0 = l * 2 * R + r, i1 = i0 + R;
      float a0r = sr[i0], a0i = si[i0], a1r = sr[i1], a1i = si[i1];
      sr[i0] = c * a0r + s * a1i;  si[i0] = c * a0i - s * a1r;
      sr[i1] = c * a1r + s * a0i;  si[i1] = c * a1i - s * a0r;
    }
  }
}

template<int Q>
__device__ __forceinline__ void apply_ry(float* sr, float* si, float c, float s) {
  constexpr int R = 1 << (5 - Q);
  constexpr int L = 1 << Q;
#pragma unroll
  for (int l = 0; l < L; ++l) {
#pragma unroll
    for (int r = 0; r < R; ++r) {
      const int i0 = l * 2 * R + r, i1 = i0 + R;
      float a0r = sr[i0], a0i = si[i0], a1r = sr[i1], a1i = si[i1];
      sr[i0] = c * a0r - s * a1r;  si[i0] = c * a0i - s * a1i;
      sr[i1] = s * a0r + c * a1r;  si[i1] = s * a0i + c * a1i;
    }
  }
}

template<int Q>
__device__ __forceinline__ void apply_rz(float* sr, float* si, float c, float s) {
  constexpr int R = 1 << (5 - Q);
  constexpr int L = 1 << Q;
#pragma unroll
  for (int l = 0; l < L; ++l) {
#pragma unroll
    for (int r = 0; r < R; ++r) {
      const int i0 = l * 2 * R + r, i1 = i0 + R;
      float a0r = sr[i0], a0i = si[i0], a1r = sr[i1], a1i = si[i1];
      sr[i0] = c * a0r + s * a0i;  si[i0] = c * a0i - s * a0r;   // * e^{-i t/2}
      sr[i1] = c * a1r - s * a1i;  si[i1] = c * a1i + s * a1r;   // * e^{+i t/2}
    }
  }
}

template<int Q>
__device__ __forceinline__ void apply_qubit(float* sr, float* si, const float* w3) {
  float c, s;
  c = cosf(0.5f * w3[0]); s = sinf(0.5f * w3[0]); apply_rx<Q>(sr, si, c, s);
  c = cosf(0.5f * w3[1]); s = sinf(0.5f * w3[1]); apply_ry<Q>(sr, si, c, s);
  c = cosf(0.5f * w3[2]); s = sinf(0.5f * w3[2]); apply_rz<Q>(sr, si, c, s);
}

__device__ __forceinline__ void apply_cz_ring(float* sr, float* si) {
  // product of CZ masks for pairs (0,1)..(4,5),(5,0); qubit 0 = MSB of idx
#pragma unroll
  for (int idx = 0; idx < DIM; ++idx) {
    int cnt = 0;
#pragma unroll
    for (int q = 0; q < NQ; ++q) {
      int b0 = (idx >> (5 - q)) & 1;
      int b1 = (idx >> (5 - ((q + 1) % NQ))) & 1;
      cnt += (b0 & b1);
    }
    if (cnt & 1) { sr[idx] = -sr[idx]; si[idx] = -si[idx]; }
  }
}

__global__ void build_unitary(const float* __restrict__ weights,
                              float* __restrict__ Mr, float* __restrict__ Mi) {
  const int j = threadIdx.x;          // basis index 0..63
  float sr[DIM], si[DIM];
#pragma unroll
  for (int i = 0; i < DIM; ++i) { sr[i] = (i == j) ? 1.0f : 0.0f; si[i] = 0.0f; }

#pragma unroll
  for (int layer = 0; layer < 2; ++layer) {
    const float* w = weights + layer * (NQ * 3);
    apply_qubit<0>(sr, si, w + 0);
    apply_qubit<1>(sr, si, w + 3);
    apply_qubit<2>(sr, si, w + 6);
    apply_qubit<3>(sr, si, w + 9);
    apply_qubit<4>(sr, si, w + 12);
    apply_qubit<5>(sr, si, w + 15);
    apply_cz_ring(sr, si);
  }
#pragma unroll
  for (int i = 0; i < DIM; ++i) {
    Mr[j * DIM + i] = sr[i];
    Mi[j * DIM + i] = si[i];
  }
}

// ------------------------------------------------------------------
// Main kernel: encoding -> E (LDS) -> psi = E @ (Mr, Mi) via WMMA ->
// probs -> Z expectations -> MLP.
// ------------------------------------------------------------------

#define ESTR 68   // LDS row stride (floats); 68 => conflict-free frag reads

__global__ void qnn_main(const float* __restrict__ x,
                         const float* __restrict__ Mr,
                         const float* __restrict__ Mi,
                         const float* __restrict__ W1,
                         const float* __restrict__ b1,
                         const float* __restrict__ W2,
                         const float* __restrict__ b2,
                         float* __restrict__ out, int B) {
  __shared__ float lds[BS * ESTR];

  const int tid = threadIdx.x;
  const int b   = blockIdx.x * BS + tid;
  const int bb  = (b < B) ? b : (B - 1);    // clamp, keeps EXEC uniform

  // ---- phase 1: encoded product state e[64] -> LDS ----
  {
    const float* xp = x + (size_t)bb * NQ;
    float cw[NQ], sw[NQ];
#pragma unroll
    for (int q = 0; q < NQ; ++q) {
      float h = 0.5f * xp[q];
      cw[q] = cosf(h);
      sw[q] = sinf(h);
    }
    float t2[2], t4[4], t8[8], t16[16], t32[32], e[64];
    t2[0] = cw[0]; t2[1] = sw[0];
#pragma unroll
    for (int i = 0; i < 2; ++i)  { t4[2*i]  = t2[i]*cw[1];  t4[2*i+1]  = t2[i]*sw[1]; }
#pragma unroll
    for (int i = 0; i < 4; ++i)  { t8[2*i]  = t4[i]*cw[2];  t8[2*i+1]  = t4[i]*sw[2]; }
#pragma unroll
    for (int i = 0; i < 8; ++i)  { t16[2*i] = t8[i]*cw[3];  t16[2*i+1] = t8[i]*sw[3]; }
#pragma unroll
    for (int i = 0; i < 16; ++i) { t32[2*i] = t16[i]*cw[4]; t32[2*i+1] = t16[i]*sw[4]; }
#pragma unroll
    for (int i = 0; i < 32; ++i) { e[2*i]   = t32[i]*cw[5]; e[2*i+1]   = t32[i]*sw[5]; }
#pragma unroll
    for (int i = 0; i < DIM; ++i) lds[tid * ESTR + i] = e[i];
  }
  __syncthreads();

  // ---- phase 2: psi = E @ M via V_WMMA_F32_16X16X4_F32, p = |psi|^2 ----
  const int lane = tid & 31;
  const int wv   = tid >> 5;            // wave id 0..3 -> N-tile
  const int lrow = lane & 15;
  const int hi   = lane >> 4;           // 0 for lanes 0-15, 1 for 16-31
  const int col  = wv * 16 + lrow;      // N column (and P column)

  // preload all B fragments for this wave's N-tile (K = 0..63 in steps of 4)
  v2f br[16], bi[16];
#pragma unroll
  for (int k = 0; k < 16; ++k) {
    const int krow = 4 * k + 2 * hi;    // lanes 0-15: K=4k,4k+1 ; 16-31: 4k+2,4k+3
    br[k].x = Mr[krow * DIM + col];
    br[k].y = Mr[(krow + 1) * DIM + col];
    bi[k].x = Mi[krow * DIM + col];
    bi[k].y = Mi[(krow + 1) * DIM + col];
  }

  float p[8][8];
#pragma unroll
  for (int m = 0; m < 8; ++m) {
    v8f accr = {};
    v8f acci = {};
    const int arow = m * 16 + lrow;
#pragma unroll
    for (int k = 0; k < 16; ++k) {
      v2f a;
      const int kb = 4 * k + 2 * hi;
      a.x = lds[arow * ESTR + kb];
      a.y = lds[arow * ESTR + kb + 1];
      accr = __builtin_amdgcn_wmma_f32_16x16x4_f32(false, a, false, br[k],
                                                   (short)0, accr, false, false);
      acci = __builtin_amdgcn_wmma_f32_16x16x4_f32(false, a, false, bi[k],
                                                   (short)0, acci, false, false);
    }
#pragma unroll
    for (int j = 0; j < 8; ++j) p[m][j] = accr[j] * accr[j] + acci[j] * acci[j];
  }
  __syncthreads();            // all waves done reading E

  // scatter probabilities into LDS (D layout: M = j + 8*hi, N = col)
#pragma unroll
  for (int m = 0; m < 8; ++m) {
#pragma unroll
    for (int j = 0; j < 8; ++j) {
      const int sample = m * 16 + j + hi * 8;
      lds[sample * ESTR + col] = p[m][j];
    }
  }
  __syncthreads();

  // ---- phase 3: Z expectations + MLP epilogue ----
  float z[NQ] = {0.f, 0.f, 0.f, 0.f, 0.f, 0.f};
#pragma unroll
  for (int idx = 0; idx < DIM; ++idx) {
    const float pp = lds[tid * ESTR + idx];
#pragma unroll
    for (int q = 0; q < NQ; ++q) {
      if ((idx >> (5 - q)) & 1) z[q] -= pp; else z[q] += pp;
    }
  }

  float o0 = b2[0], o1 = b2[1];
#pragma unroll
  for (int j = 0; j < 32; ++j) {
    float h = b1[j];
#pragma unroll
    for (int q = 0; q < NQ; ++q) h = fmaf(W1[j * NQ + q], z[q], h);
    h = fmaxf(h, 0.0f);
    o0 = fmaf(W2[j], h, o0);
    o1 = fmaf(W2[32 + j], h, o1);
  }

  if (b < B) {
    out[(size_t)b * 2 + 0] = o0;
    out[(size_t)b * 2 + 1] = o1;
  }
}

// ------------------------------------------------------------------

extern "C" void kernel_launch(void* const* d_in, const int* in_sizes, int n_in,
                              void* d_out, int out_size, void* d_ws, size_t ws_size,
                              hipStream_t stream) {
  const float* x  = (const float*)d_in[0];
  const float* w  = (const float*)d_in[1];
  const float* W1 = (const float*)d_in[2];
  const float* b1 = (const float*)d_in[3];
  const float* W2 = (const float*)d_in[4];
  const float* b2 = (const float*)d_in[5];
  float* out = (float*)d_out;

  float* Mr = (float*)d_ws;           // 64*64 floats
  float* Mi = Mr + DIM * DIM;         // 64*64 floats (32 KB total)

  const int B = in_sizes[0] / NQ;     // 262144

  build_unitary<<<1, DIM, 0, stream>>>(w, Mr, Mi);

  const int blocks = (B + BS - 1) / BS;
  qnn_main<<<blocks, BS, 0, stream>>>(x, Mr, Mi, W1, b1, W2, b2, out, B);
}